// MappingNetwork_39152921870485
// MI455X (gfx1250) — compile-verified
//
#include <hip/hip_runtime.h>
#include <hip/hip_bf16.h>

// Problem constants (match reference)
#define BATCH   65536
#define NEXP    16
#define HDIM    128
#define DDIM    64
#define ZDIM    16
#define LDS_STRIDE 132                    // 128 + 4 pad -> conflict-free m*132+k access
#define ROWS    64                        // rows (samples) per block
#define MTILES  (ROWS / 16)               // 4 independent 16x16 accum tiles per wave
#define NTILES_MAX (BATCH / ROWS + NEXP)  // 1040: worst-case blocks incl. per-expert tail pad

typedef __attribute__((ext_vector_type(2))) float v2f;
typedef __attribute__((ext_vector_type(8))) float v8f;

__device__ __forceinline__ v8f wmma_f32(v2f a, v2f b, v8f c) {
  // D = A(16x4) * B(4x16) + C(16x16), full f32
  return __builtin_amdgcn_wmma_f32_16x16x4_f32(
      /*neg_a=*/false, a, /*neg_b=*/false, b,
      /*c_mod=*/(short)0, c, /*reuse_a=*/false, /*reuse_b=*/false);
}

// ---------------- prep kernels ----------------

__global__ __launch_bounds__(64) void k_zero(int* p) {
  if (threadIdx.x < 32) p[threadIdx.x] = 0;   // counts[16] + cursors[16]
}

__global__ __launch_bounds__(256) void k_hist(const int* __restrict__ y,
                                              int* __restrict__ counts) {
  int b = blockIdx.x * 256 + threadIdx.x;
  if (b < BATCH) atomicAdd(&counts[y[b] & (NEXP - 1)], 1);
}

__global__ __launch_bounds__(256) void k_scatter(const int* __restrict__ y,
                                                 const int* __restrict__ counts,
                                                 int* __restrict__ cursors,
                                                 int* __restrict__ perm) {
  __shared__ int soff[NEXP];
  if (threadIdx.x == 0) {
    int o = 0;
    for (int e = 0; e < NEXP; ++e) { soff[e] = o; o += counts[e]; }
  }
  __syncthreads();
  int b = blockIdx.x * 256 + threadIdx.x;
  if (b < BATCH) {
    int e = y[b] & (NEXP - 1);
    int pos = soff[e] + atomicAdd(&cursors[e], 1);
    perm[pos] = b;   // bucket order is nondeterministic, but out[] scatter by sample id is not
  }
}

// ---------------- main compute ----------------

// Per wave: 4 stacked 16x16 output tiles (rows t*16..t*16+15, cols n0..n0+15):
//   C_t += X[ROWS, Kdim] * W[Kdim, n0..n0+16)
// B-operand loaded once per k-step and shared by 4 independent WMMA chains.
// src: LDS row-major [ROWS][LDS_STRIDE]; W: global row-major, leading dim ldw.
__device__ __forceinline__ void layer_tile(const float* __restrict__ src,
                                           const float* __restrict__ W, int ldw,
                                           const float* __restrict__ bias,
                                           float* __restrict__ dst,
                                           int Kdim, int lane, int wave) {
  const int n0   = wave << 4;
  const int m    = lane & 15;            // A row within sub-tile / B,C column within slice
  const int koff = (lane >> 4) << 1;     // lanes 16-31 hold K+2, K+3
  const int n    = n0 + m;               // output column
  v8f c0 = {}, c1 = {}, c2 = {}, c3 = {};
  const float* a_ptr = src + m * LDS_STRIDE + koff;
  const float* b_ptr = W + (size_t)koff * ldw + n;
  const int steps = Kdim >> 2;
  #pragma unroll 4
  for (int kk = 0; kk < steps; ++kk) {
    v2f b; b.x = b_ptr[0]; b.y = b_ptr[ldw];                   // W[k][n], W[k+1][n]
    v2f a0; a0.x = a_ptr[0 * 16 * LDS_STRIDE];     a0.y = a_ptr[0 * 16 * LDS_STRIDE + 1];
    v2f a1; a1.x = a_ptr[1 * 16 * LDS_STRIDE];     a1.y = a_ptr[1 * 16 * LDS_STRIDE + 1];
    v2f a2; a2.x = a_ptr[2 * 16 * LDS_STRIDE];     a2.y = a_ptr[2 * 16 * LDS_STRIDE + 1];
    v2f a3; a3.x = a_ptr[3 * 16 * LDS_STRIDE];     a3.y = a_ptr[3 * 16 * LDS_STRIDE + 1];
    c0 = wmma_f32(a0, b, c0);            // 4 independent accumulation chains
    c1 = wmma_f32(a1, b, c1);
    c2 = wmma_f32(a2, b, c2);
    c3 = wmma_f32(a3, b, c3);
    a_ptr += 4;
    b_ptr += 4 * (size_t)ldw;
  }
  const float bv = bias[n];              // same N for all accum regs
  const int mbase = (lane >> 4) << 3;    // lanes 16-31 hold rows M=8..15 of each sub-tile
  v8f acc[4] = {c0, c1, c2, c3};
  #pragma unroll
  for (int t = 0; t < MTILES; ++t) {
    #pragma unroll
    for (int v = 0; v < 8; ++v) {
      float x = acc[t][v] + bv;
      x = fmaxf(x, 0.0f);
      dst[(t * 16 + mbase + v) * LDS_STRIDE + n] = x;
    }
  }
}

// Final layer: 128 -> 64, no ReLU, scatter rows to out[gid][n]
__device__ __forceinline__ void layer_out(const float* __restrict__ src,
                                          const float* __restrict__ W,
                                          const float* __restrict__ bias,
                                          float* __restrict__ out,
                                          const int* __restrict__ sh_gid,
                                          int nvalid, int lane, int wave) {
  if (wave >= 4) return;                 // D=64 -> 4 column tiles (wave-uniform branch)
  const int m    = lane & 15;
  const int koff = (lane >> 4) << 1;
  const int n    = (wave << 4) + m;
  v8f c0 = {}, c1 = {}, c2 = {}, c3 = {};
  const float* a_ptr = src + m * LDS_STRIDE + koff;
  const float* b_ptr = W + (size_t)koff * DDIM + n;
  #pragma unroll 4
  for (int kk = 0; kk < HDIM / 4; ++kk) {
    v2f b; b.x = b_ptr[0]; b.y = b_ptr[DDIM];
    v2f a0; a0.x = a_ptr[0 * 16 * LDS_STRIDE];     a0.y = a_ptr[0 * 16 * LDS_STRIDE + 1];
    v2f a1; a1.x = a_ptr[1 * 16 * LDS_STRIDE];     a1.y = a_ptr[1 * 16 * LDS_STRIDE + 1];
    v2f a2; a2.x = a_ptr[2 * 16 * LDS_STRIDE];     a2.y = a_ptr[2 * 16 * LDS_STRIDE + 1];
    v2f a3; a3.x = a_ptr[3 * 16 * LDS_STRIDE];     a3.y = a_ptr[3 * 16 * LDS_STRIDE + 1];
    c0 = wmma_f32(a0, b, c0);
    c1 = wmma_f32(a1, b, c1);
    c2 = wmma_f32(a2, b, c2);
    c3 = wmma_f32(a3, b, c3);
    a_ptr += 4;
    b_ptr += 4 * DDIM;
  }
  const float bv = bias[n];
  const int mbase = (lane >> 4) << 3;
  v8f acc[4] = {c0, c1, c2, c3};
  #pragma unroll
  for (int t = 0; t < MTILES; ++t) {
    #pragma unroll
    for (int v = 0; v < 8; ++v) {
      int mm = t * 16 + mbase + v;
      if (mm < nvalid) {
        out[(size_t)sh_gid[mm] * DDIM + n] = acc[t][v] + bv;
      }
    }
  }
}

__global__ __launch_bounds__(256) void mapping_main(
    const float* __restrict__ z,
    const float* __restrict__ bw0, const float* __restrict__ bb0,
    const float* __restrict__ bw1, const float* __restrict__ bb1,
    const float* __restrict__ bw2, const float* __restrict__ bb2,
    const float* __restrict__ bw3, const float* __restrict__ bb3,
    const float* __restrict__ hw0, const float* __restrict__ hb0,
    const float* __restrict__ hw1, const float* __restrict__ hb1,
    const float* __restrict__ hw2, const float* __restrict__ hb2,
    const float* __restrict__ hw3, const float* __restrict__ hb3,
    const int* __restrict__ counts, const int* __restrict__ perm,
    float* __restrict__ out) {
  __shared__ float actA[ROWS * LDS_STRIDE];
  __shared__ float actB[ROWS * LDS_STRIDE];
  __shared__ int sh_meta[3];     // expert, start, end
  __shared__ int sh_gid[ROWS];

  const int tid  = threadIdx.x;
  const int lane = tid & 31;
  const int wave = tid >> 5;

  // Map block -> (expert, row range) from bucket histogram (cheap 16-step scan per block).
  if (tid == 0) {
    int t = blockIdx.x;
    int off = 0, base = 0, expert = -1, start = 0, end = 0;
    for (int e = 0; e < NEXP; ++e) {
      int cnt = counts[e];
      int nt  = (cnt + ROWS - 1) / ROWS;
      if (expert < 0 && t < base + nt) {
        expert = e;
        start  = off + (t - base) * ROWS;
        end    = off + cnt;
      }
      base += nt;
      off  += cnt;
    }
    sh_meta[0] = expert; sh_meta[1] = start; sh_meta[2] = end;
  }
  __syncthreads();
  const int expert = sh_meta[0];
  if (expert < 0) return;        // past the last real tile (whole block exits)
  const int start = sh_meta[1];
  const int end   = sh_meta[2];
  const int nvalid = min(ROWS, end - start);

  // Load z tile [ROWS][16] into LDS (tail rows clamped to a valid sample; recomputed, never stored)
  {
    #pragma unroll
    for (int i = 0; i < ROWS * ZDIM / 256; ++i) {
      int idx = i * 256 + tid;
      int r = idx >> 4, cidx = idx & 15;
      int row = start + r;
      if (row >= end) row = end - 1;
      int g = perm[row];
      if (cidx == 0) sh_gid[r] = g;
      actA[r * LDS_STRIDE + cidx] = z[(size_t)g * ZDIM + cidx];
    }
  }
  __syncthreads();

  // Backbone: 4 x (Linear + ReLU)
  layer_tile(actA, bw0, HDIM, bb0, actB, ZDIM, lane, wave);  __syncthreads();
  layer_tile(actB, bw1, HDIM, bb1, actA, HDIM, lane, wave);  __syncthreads();
  layer_tile(actA, bw2, HDIM, bb2, actB, HDIM, lane, wave);  __syncthreads();
  layer_tile(actB, bw3, HDIM, bb3, actA, HDIM, lane, wave);  __syncthreads();

  // Selected expert head only (mathematically identical to compute-all + gather)
  const size_t we = (size_t)expert * HDIM * HDIM;
  layer_tile(actA, hw0 + we, HDIM, hb0 + expert * HDIM, actB, HDIM, lane, wave); __syncthreads();
  layer_tile(actB, hw1 + we, HDIM, hb1 + expert * HDIM, actA, HDIM, lane, wave); __syncthreads();
  layer_tile(actA, hw2 + we, HDIM, hb2 + expert * HDIM, actB, HDIM, lane, wave); __syncthreads();
  layer_out(actB, hw3 + (size_t)expert * HDIM * DDIM, hb3 + expert * DDIM,
            out, sh_gid, nvalid, lane, wave);
}

// ---------------- launch ----------------

extern "C" void kernel_launch(void* const* d_in, const int* in_sizes, int n_in,
                              void* d_out, int out_size, void* d_ws, size_t ws_size,
                              hipStream_t stream) {
  const float* z   = (const float*)d_in[0];
  const int*   y   = (const int*)  d_in[1];
  const float* bw0 = (const float*)d_in[2];
  const float* bb0 = (const float*)d_in[3];
  const float* bw1 = (const float*)d_in[4];
  const float* bb1 = (const float*)d_in[5];
  const float* bw2 = (const float*)d_in[6];
  const float* bb2 = (const float*)d_in[7];
  const float* bw3 = (const float*)d_in[8];
  const float* bb3 = (const float*)d_in[9];
  const float* hw0 = (const float*)d_in[10];
  const float* hb0 = (const float*)d_in[11];
  const float* hw1 = (const float*)d_in[12];
  const float* hb1 = (const float*)d_in[13];
  const float* hw2 = (const float*)d_in[14];
  const float* hb2 = (const float*)d_in[15];
  const float* hw3 = (const float*)d_in[16];
  const float* hb3 = (const float*)d_in[17];
  float* out = (float*)d_out;

  // Workspace layout: counts[16] | cursors[16] | perm[BATCH]  (~256 KB)
  int* counts  = (int*)d_ws;
  int* cursors = counts + 16;
  int* perm    = counts + 32;

  k_zero<<<1, 64, 0, stream>>>(counts);
  k_hist<<<BATCH / 256, 256, 0, stream>>>(y, counts);
  k_scatter<<<BATCH / 256, 256, 0, stream>>>(y, counts, cursors, perm);
  mapping_main<<<NTILES_MAX, 256, 0, stream>>>(
      z, bw0, bb0, bw1, bb1, bw2, bb2, bw3, bb3,
      hw0, hb0, hw1, hb1, hw2, hb2, hw3, hb3,
      counts, perm, out);
}